// TorchSurvCoxLoss_26319559590547
// MI455X (gfx1250) — compile-verified
//
#include <hip/hip_runtime.h>
#include <stdint.h>

#define T_BINS 32768
#define SCAN_BLK 256
#define N_SCAN_BLKS (T_BINS / SCAN_BLK)   // 128

typedef float v2f __attribute__((ext_vector_type(2)));
typedef float v8f __attribute__((ext_vector_type(8)));

// ---- workspace layout (bytes) ----
// s[32768]f32 @0 | ties[32768]f32 @131072 | m[32768]u32 @262144 | cnt[32768]u32 @393216
// wsuf[32768]f32 @524288 | blkSums[128]f32 @655360 | blkOff[128]f32 @655872
// scal: double sumLH @656384, double sumLogDen @656392 | nbins u32 @656400
#define WS_WORDS 164101

__global__ void csx_zero(uint32_t* __restrict__ w, int nwords) {
    int i = blockIdx.x * blockDim.x + threadIdx.x;
    int stride = gridDim.x * blockDim.x;
    for (; i < nwords; i += stride) w[i] = 0u;
}

__device__ __forceinline__ void block_reduce_add_f64(float loc, double* target) {
    __shared__ float red[32];
    for (int o = 16; o > 0; o >>= 1) loc += __shfl_down(loc, o, 32);
    int lane = threadIdx.x & 31, wid = threadIdx.x >> 5;
    if (lane == 0) red[wid] = loc;
    __syncthreads();
    if (wid == 0) {
        int nw = blockDim.x >> 5;
        float v = (lane < nw) ? red[lane] : 0.0f;
        for (int o = 16; o > 0; o >>= 1) v += __shfl_down(v, o, 32);
        if (lane == 0) atomicAdd(target, (double)v);
    }
}

// Pass 1: per-bin aggregates via L2-resident atomics + global scalar sum of lh over events.
__global__ void csx_bin(const float* __restrict__ lh, const uint8_t* __restrict__ ev,
                        const int* __restrict__ tt,
                        float* __restrict__ s, float* __restrict__ ties,
                        uint32_t* __restrict__ m_int, double* __restrict__ sumLH, int N) {
    int i = blockIdx.x * blockDim.x + threadIdx.x;
    int stride = gridDim.x * blockDim.x;
    float loc = 0.0f;
    for (; i < N; i += stride) {
        float x = lh[i];
        int b = tt[i];
        float e = expf(x);
        atomicAdd(&s[b], e);
        if (ev[i]) {
            atomicAdd(&ties[b], e);
            atomicAdd(&m_int[b], 1u);
            loc += x;
        }
    }
    block_reduce_add_f64(loc, sumLH);
}

// Pass 2: inclusive suffix-sum of s per 256-bin block via V_WMMA_F32_16X16X4_F32.
// One wave per block. X[r][c] = s[base + c*16 + r]; Y = U16 * X (U upper-tri incl diag)
// accumulated over 4 K-chunks; cross-column carry via lane shuffles.
__global__ void csx_scan(const float* __restrict__ s, const uint32_t* __restrict__ m_int,
                         float* __restrict__ wsuf, float* __restrict__ blkSums,
                         uint32_t* __restrict__ nbins) {
    const int blk  = blockIdx.x;          // 0..127, blockDim.x == 32
    const int base = blk * SCAN_BLK;
    const int lane = threadIdx.x;
    const int n  = lane & 15;             // B/D column
    const int kh = lane >> 4;             // lane-half select
    const int mrow = lane & 15;           // A row

    v8f acc = {};
#pragma unroll
    for (int c4 = 0; c4 < 4; ++c4) {
        // ISA A layout (16x4 f32): v0 holds K=2*kh, v1 holds K=2*kh+1, M = lane&15
        int k0 = 4 * c4 + 2 * kh;
        v2f a, b;
        a.x = (k0     >= mrow) ? 1.0f : 0.0f;   // U[m][k] = (k >= m)
        a.y = (k0 + 1 >= mrow) ? 1.0f : 0.0f;
        // B (4x16 f32): same K split as A; row k of chunk = X[4*c4 + k][n]
        b.x = s[base + n * 16 + k0];
        b.y = s[base + n * 16 + k0 + 1];
        acc = __builtin_amdgcn_wmma_f32_16x16x4_f32(
            /*neg_a=*/false, a, /*neg_b=*/false, b,
            /*c_mod=*/(short)0, acc, /*reuse_a=*/false, /*reuse_b=*/false);
    }

    // Column totals = Y[0][c] live in acc[0] on lanes 0..15 (M=0).
    float ct = acc[0];
    float T = 0.0f;                 // carry from later columns: sum_{c' > n} total(c')
#pragma unroll
    for (int cp = 1; cp < 16; ++cp) {
        float v = __shfl(ct, cp, 32);
        if (cp > n) T += v;
    }

    // D layout: M = v + 8*kh, N = n; element index = n*16 + M.
#pragma unroll
    for (int v = 0; v < 8; ++v)
        wsuf[base + n * 16 + v + 8 * kh] = acc[v] + T;

    if (lane == 0) blkSums[blk] = acc[0] + T;   // total of this 256-bin block

    // Count non-empty bins in this block.
    uint32_t c = 0;
#pragma unroll
    for (int j = 0; j < 8; ++j)
        c += (m_int[base + lane * 8 + j] != 0u) ? 1u : 0u;
    for (int o = 16; o > 0; o >>= 1) c += __shfl_down(c, o, 32);
    if (lane == 0) atomicAdd(nbins, c);
}

// Exclusive suffix over the 128 block totals (tiny).
__global__ void csx_blkoff(const float* __restrict__ blkSums, float* __restrict__ blkOff) {
    int b = threadIdx.x;              // blockDim.x == 128
    float t = 0.0f;
    for (int bp = b + 1; bp < N_SCAN_BLKS; ++bp) t += blkSums[bp];
    blkOff[b] = t;
}

// Pass 3: per-event Efron log terms; ticket counter assigns ranks 0..m-1 (order-free sum).
__global__ void csx_efron(const float* __restrict__ lh, const uint8_t* __restrict__ ev,
                          const int* __restrict__ tt,
                          const float* __restrict__ ties, const uint32_t* __restrict__ m_int,
                          uint32_t* __restrict__ cnt,
                          const float* __restrict__ wsuf, const float* __restrict__ blkOff,
                          double* __restrict__ sumLogDen, int N) {
    int i = blockIdx.x * blockDim.x + threadIdx.x;
    int stride = gridDim.x * blockDim.x;
    float loc = 0.0f;
    for (; i < N; i += stride) {
        if (ev[i]) {
            int b = tt[i];
            uint32_t l = atomicAdd(&cnt[b], 1u);
            float D  = wsuf[b] + blkOff[b >> 8];
            float Dt = ties[b];
            float mi = (float)m_int[b];
            loc += logf(D - ((float)l / mi) * Dt);
        }
        (void)lh;
    }
    block_reduce_add_f64(loc, sumLogDen);
}

__global__ void csx_final(const double* __restrict__ scal, const uint32_t* __restrict__ nbins,
                          float* __restrict__ out) {
    // loss = (sum log_den - sum log_nom) / #included bins
    out[0] = (float)((scal[1] - scal[0]) / (double)(*nbins));
}

extern "C" void kernel_launch(void* const* d_in, const int* in_sizes, int n_in,
                              void* d_out, int out_size, void* d_ws, size_t ws_size,
                              hipStream_t stream) {
    const float*   lh = (const float*)d_in[0];
    const uint8_t* ev = (const uint8_t*)d_in[1];   // jnp bool -> 1 byte/elem
    const int*     tt = (const int*)d_in[2];
    int N = in_sizes[0];

    char* ws = (char*)d_ws;
    float*    s       = (float*)(ws + 0);
    float*    ties    = (float*)(ws + 131072);
    uint32_t* m_int   = (uint32_t*)(ws + 262144);
    uint32_t* cnt     = (uint32_t*)(ws + 393216);
    float*    wsuf    = (float*)(ws + 524288);
    float*    blkSums = (float*)(ws + 655360);
    float*    blkOff  = (float*)(ws + 655872);
    double*   scal    = (double*)(ws + 656384);    // [0]=sumLH [1]=sumLogDen
    uint32_t* nbins   = (uint32_t*)(ws + 656400);

    csx_zero<<<256, 256, 0, stream>>>((uint32_t*)ws, WS_WORDS);
    csx_bin<<<8192, 256, 0, stream>>>(lh, ev, tt, s, ties, m_int, &scal[0], N);
    csx_scan<<<N_SCAN_BLKS, 32, 0, stream>>>(s, m_int, wsuf, blkSums, nbins);
    csx_blkoff<<<1, N_SCAN_BLKS, 0, stream>>>(blkSums, blkOff);
    csx_efron<<<8192, 256, 0, stream>>>(lh, ev, tt, ties, m_int, cnt, wsuf, blkOff,
                                        &scal[1], N);
    csx_final<<<1, 1, 0, stream>>>(scal, nbins, (float*)d_out);
}